// MoeRouter_73641509257561
// MI455X (gfx1250) — compile-verified
//
#include <hip/hip_runtime.h>
#include <hip/hip_bf16.h>

typedef __attribute__((ext_vector_type(2))) float v2f;
typedef __attribute__((ext_vector_type(8))) float v8f;

#define HIDDEN 2048
#define NEXP 8
#define TOKENS_PER_BLOCK 128
#define THREADS 256

__global__ __launch_bounds__(THREADS) void moe_router_wmma(
    const float* __restrict__ x,      // [N, 2048]
    const float* __restrict__ w,      // [8, 2048]
    const float* __restrict__ bias,   // [8]
    float* __restrict__ out_w,        // [N, 8] fp32
    int*   __restrict__ out_i,        // [N, 8] int32
    int n_tokens)
{
    // Weight staged as K-pairs: wT2[p][n] = (w[n][2p], w[n][2p+1]), n in 0..7.
    // 1024 * 8 * 8B = 64 KB. Reused as score scratch after the GEMM phase.
    __shared__ v2f wT2[(HIDDEN / 2) * NEXP];

    const int tid = threadIdx.x;

    // ---- Stage weights into LDS (once per block; weight is L2-hot) ----
    for (int i = tid; i < (HIDDEN / 2) * NEXP; i += THREADS) {
        int p = i >> 3;        // k-pair index 0..1023
        int n = i & 7;         // expert
        v2f v;
        v.x = w[n * HIDDEN + 2 * p];
        v.y = w[n * HIDDEN + 2 * p + 1];
        wT2[i] = v;
    }
    __syncthreads();

    const int wave = tid >> 5;   // wave32: 8 waves per block
    const int lane = tid & 31;
    const int rown = lane & 15;  // A: token row / B,C,D: column (expert)
    const int hi   = lane >> 4;  // upper half-wave -> K offset +2

    const int tile_base = blockIdx.x * TOKENS_PER_BLOCK + wave * 16;
    const float* xrow = x + (size_t)(tile_base + rown) * HIDDEN + 2 * hi;

    // Zero-mask for B columns 8..15 (N padded to 16)
    const float bmsk = (rown < NEXP) ? 1.0f : 0.0f;

    v8f acc0 = {};
    v8f acc1 = {};
    #pragma unroll 4
    for (int k = 0; k < HIDDEN; k += 8) {
        // A fragment: lane holds x[row][k + koff .. k + koff + 1]
        v2f a0 = *(const v2f*)(xrow + k);
        v2f b0 = wT2[((k >> 1) + hi) * NEXP + (rown & 7)] * bmsk;
        acc0 = __builtin_amdgcn_wmma_f32_16x16x4_f32(
            false, a0, false, b0, (short)0, acc0, false, false);

        v2f a1 = *(const v2f*)(xrow + k + 4);
        v2f b1 = wT2[(((k + 4) >> 1) + hi) * NEXP + (rown & 7)] * bmsk;
        acc1 = __builtin_amdgcn_wmma_f32_16x16x4_f32(
            false, a1, false, b1, (short)0, acc1, false, false);
    }
    v8f acc = acc0 + acc1;

    // ---- Transpose scores via LDS (reuse wT2 region as [wave][token16][8]) ----
    __syncthreads();   // everyone done reading wT2 as weights
    float* scr = reinterpret_cast<float*>(wT2);

    // D layout: lanes 0-15 hold (M=r, N=lane) in VGPR r; lanes 16-31 hold M=r+8.
    if (rown < NEXP) {
        #pragma unroll
        for (int r = 0; r < 8; ++r) {
            int m = hi ? (r + 8) : r;                 // token row within tile
            scr[(wave * 16 + m) * NEXP + rown] = acc[r];
        }
    }
    __syncthreads();

    // ---- Per-token sigmoid + full descending argsort + normalize ----
    if (lane < 16) {
        const int m = lane;
        const int token = tile_base + m;

        float key[NEXP];
        int   id[NEXP];
        float sum = 0.0f;
        #pragma unroll
        for (int e = 0; e < NEXP; ++e) {
            float z  = scr[(wave * 16 + m) * NEXP + e];
            float sg = 1.0f / (1.0f + __expf(-z));
            sum += sg;
            key[e] = sg + bias[e];   // bias-corrected selection score
            id[e]  = e;
        }

        // Stable descending sort: swap when key[a] < key[b], ties keep lower
        // index first (matches jax.lax.top_k tie-breaking).
        auto CSWAP = [&](int a, int b) {
            bool sw = (key[a] < key[b]) ||
                      ((key[a] == key[b]) && (id[a] > id[b]));
            float ka = sw ? key[b] : key[a];
            float kb = sw ? key[a] : key[b];
            int   ia = sw ? id[b]  : id[a];
            int   ib = sw ? id[a]  : id[b];
            key[a] = ka; key[b] = kb; id[a] = ia; id[b] = ib;
        };
        // Batcher odd-even merge network for n=8 (19 comparators)
        CSWAP(0,1); CSWAP(2,3); CSWAP(4,5); CSWAP(6,7);
        CSWAP(0,2); CSWAP(1,3); CSWAP(4,6); CSWAP(5,7);
        CSWAP(1,2); CSWAP(5,6);
        CSWAP(0,4); CSWAP(1,5); CSWAP(2,6); CSWAP(3,7);
        CSWAP(2,4); CSWAP(3,5);
        CSWAP(1,2); CSWAP(3,4); CSWAP(5,6);

        const float inv = 1.4f / (sum + 1e-10f);
        #pragma unroll
        for (int j = 0; j < NEXP; ++j) {
            float sg = key[j] - bias[id[j]];  // recover raw sigmoid score
            out_w[(size_t)token * NEXP + j] = sg * inv;
            out_i[(size_t)token * NEXP + j] = id[j];
        }
    }
}

extern "C" void kernel_launch(void* const* d_in, const int* in_sizes, int n_in,
                              void* d_out, int out_size, void* d_ws, size_t ws_size,
                              hipStream_t stream) {
    const float* x    = (const float*)d_in[0];   // hidden_states [4,8192,2048] f32
    const float* w    = (const float*)d_in[1];   // weight [8,2048] f32
    const float* bias = (const float*)d_in[2];   // e_score_correction_bias [8] f32

    const int n_tokens = in_sizes[0] / HIDDEN;   // 32768

    float* out_w = (float*)d_out;                                  // [N,8] f32
    int*   out_i = (int*)((float*)d_out + (size_t)n_tokens * NEXP); // [N,8] i32

    const int blocks = n_tokens / TOKENS_PER_BLOCK;                // 256
    if (blocks > 0) {
        moe_router_wmma<<<blocks, THREADS, 0, stream>>>(
            x, w, bias, out_w, out_i, n_tokens);
    }
}